// _PCL_Losses_57964878627195
// MI455X (gfx1250) — compile-verified
//
#include <hip/hip_runtime.h>
#include <hip/hip_bf16.h>

typedef __attribute__((ext_vector_type(2))) float v2f;
typedef __attribute__((ext_vector_type(8))) float v8f;

#define NBLK_BG 240
#define NBLK_FG 16
#define NPARTS  (NBLK_BG + NBLK_FG)   // 256, matches finalize blockDim
#define BLK     256

// Reduce (acc0, acc1) across the whole block, in pure f32, using the CDNA5
// f32 WMMA as the cross-lane reducer:
//   A (16x4 f32, 2 VGPRs/lane) holds the 64 per-lane partials of this wave,
//   B (4x16 f32) = all ones  =>  D[m][n] = rowsum(A[m]).
// Column n of D: lanes L (M=0..7 in d[0..7]) and L+16 (M=8..15). So
// sum(d[0..7]) + shfl_down(.,16) = exact f32 sum of all 64 A entries.
// EXEC must be all ones: every thread of the block calls this (no divergence).
__device__ __forceinline__ float block_reduce_wmma(float acc0, float acc1,
                                                   float* lds) {
  v2f a; a[0] = acc0; a[1] = acc1;
  v2f b; b[0] = 1.0f; b[1] = 1.0f;
  v8f c = {};
  v8f d = __builtin_amdgcn_wmma_f32_16x16x4_f32(
      /*neg_a=*/false, a, /*neg_b=*/false, b,
      /*c_mod=*/(short)0, c, /*reuse_a=*/false, /*reuse_b=*/false);
  float s = d[0] + d[1] + d[2] + d[3] + d[4] + d[5] + d[6] + d[7];
  s += __shfl_down(s, 16, 32);          // lane L += lane L+16  (wave32)
  const int lane = threadIdx.x & 31;
  const int wave = threadIdx.x >> 5;
  if (lane == 0) lds[wave] = s;
  __syncthreads();
  float total = 0.0f;
  if (threadIdx.x == 0) {
    const int nwaves = blockDim.x >> 5;
    for (int w = 0; w < nwaves; ++w) total += lds[w];   // fixed order: deterministic
  }
  return total;   // valid on thread 0 only
}

// Background term: sum over proposals with label==0 of w[i]*log(pcl_prob[i,0]).
// Dominant traffic: strided column-0 reads of pcl_prob (stride C*4 = 324 B).
__global__ void pcl_bg_kernel(const float* __restrict__ pcl_prob,
                              const int* __restrict__ labels,
                              const float* __restrict__ w,
                              const float* __restrict__ im_labels,
                              float* __restrict__ partials,
                              int N, int C) {
  __shared__ float lds[BLK / 32];
  const bool gate = (im_labels[0] != 0.0f);
  const int stride = gridDim.x * blockDim.x;
  float acc0 = 0.0f, acc1 = 0.0f;

  for (int i = blockIdx.x * blockDim.x + threadIdx.x; i < N; i += 2 * stride) {
    {
      const int lbl = labels[i];
      // prefetch next strided column element (speculative; dropped if OOB)
      __builtin_prefetch(&pcl_prob[(size_t)(i + 2 * stride) * (size_t)C], 0, 1);
      if (gate && lbl == 0)
        acc0 += w[i] * logf(pcl_prob[(size_t)i * (size_t)C]);
    }
    const int j = i + stride;
    if (j < N) {
      const int lbl = labels[j];
      if (gate && lbl == 0)
        acc1 += w[j] * logf(pcl_prob[(size_t)j * (size_t)C]);
    }
  }
  const float total = block_reduce_wmma(acc0, acc1, lds);
  if (threadIdx.x == 0) partials[blockIdx.x] = total;
}

// Foreground term: sum over clusters with pc_labels>0 (and im_labels gate) of
// img_w[g]*log(pc_probs[g]).
__global__ void pcl_fg_kernel(const int* __restrict__ pc_labels,
                              const float* __restrict__ pc_probs,
                              const float* __restrict__ img_w,
                              const float* __restrict__ im_labels,
                              float* __restrict__ partials,
                              int G) {
  __shared__ float lds[BLK / 32];
  const int stride = gridDim.x * blockDim.x;
  float acc0 = 0.0f, acc1 = 0.0f;

  for (int g = blockIdx.x * blockDim.x + threadIdx.x; g < G; g += 2 * stride) {
    {
      const int lbl = pc_labels[g];
      if (lbl > 0 && im_labels[lbl] != 0.0f)
        acc0 += img_w[g] * logf(pc_probs[g]);
    }
    const int h = g + stride;
    if (h < G) {
      const int lbl = pc_labels[h];
      if (lbl > 0 && im_labels[lbl] != 0.0f)
        acc1 += img_w[h] * logf(pc_probs[h]);
    }
  }
  const float total = block_reduce_wmma(acc0, acc1, lds);
  if (threadIdx.x == 0) partials[NBLK_BG + blockIdx.x] = total;
}

// Final deterministic reduce of the 256 partials; out = -(bg+fg)/N.
__global__ void pcl_finalize_kernel(const float* __restrict__ partials,
                                    float* __restrict__ out, int N) {
  __shared__ float lds[BLK / 32];
  const float v = (threadIdx.x < NPARTS) ? partials[threadIdx.x] : 0.0f;
  const float total = block_reduce_wmma(v, 0.0f, lds);
  if (threadIdx.x == 0) out[0] = -total / (float)N;
}

extern "C" void kernel_launch(void* const* d_in, const int* in_sizes, int n_in,
                              void* d_out, int out_size, void* d_ws, size_t ws_size,
                              hipStream_t stream) {
  // setup_inputs order:
  // 0 pcl_prob (N*C f32), 1 labels (N i32), 2 cls_loss_weights (N f32),
  // 3 gt_assignment (unused), 4 pc_labels (G i32), 5 pc_probs (G f32),
  // 6 pc_count (unused), 7 img_cls_loss_weights (G f32), 8 im_labels_real (C f32)
  const float* pcl_prob  = (const float*)d_in[0];
  const int*   labels    = (const int*)d_in[1];
  const float* clsw      = (const float*)d_in[2];
  const int*   pc_labels = (const int*)d_in[4];
  const float* pc_probs  = (const float*)d_in[5];
  const float* img_w     = (const float*)d_in[7];
  const float* im_labels = (const float*)d_in[8];
  float* out = (float*)d_out;

  const int N = in_sizes[1];
  const int C = (N > 0) ? (in_sizes[0] / N) : 0;
  const int G = in_sizes[4];

  float* partials = (float*)d_ws;   // NPARTS floats of scratch

  pcl_bg_kernel<<<NBLK_BG, BLK, 0, stream>>>(pcl_prob, labels, clsw, im_labels,
                                             partials, N, C);
  pcl_fg_kernel<<<NBLK_FG, BLK, 0, stream>>>(pc_labels, pc_probs, img_w,
                                             im_labels, partials, G);
  pcl_finalize_kernel<<<1, BLK, 0, stream>>>(partials, out, N);
}